// SparseLayer_4226247819797
// MI455X (gfx1250) — compile-verified
//
#include <hip/hip_runtime.h>

typedef float v2f __attribute__((ext_vector_type(2)));
typedef float v8f __attribute__((ext_vector_type(8)));

#define IN_DIM   8192
#define OUT_DIM  4096
#define BATCH    4096
#define NNZ_CNT  1638400

#define BM 128
#define BN 128
#define KB 32          // K per LDS stage -> 16 packed K-pairs
#define ROWP 144       // float2 per LDS row (16 pad): 288 dwords == 32 mod 64
                       //  -> half-waves (rows k2 / k2+1) hit disjoint bank halves

// ---------------------------------------------------------------------------
// Phase 1: zero the dense weight matrix in workspace (128 MB)
// ---------------------------------------------------------------------------
__global__ void zero_w_kernel(float4* __restrict__ W4, int n4) {
    int i = blockIdx.x * blockDim.x + threadIdx.x;
    if (i < n4) W4[i] = make_float4(0.f, 0.f, 0.f, 0.f);
}

// ---------------------------------------------------------------------------
// Phase 2: COO scatter-add into dense W (duplicates accumulate -> atomics)
// ---------------------------------------------------------------------------
__global__ void scatter_w_kernel(const int* __restrict__ row_idx,
                                 const int* __restrict__ col_idx,
                                 const float* __restrict__ wv,
                                 float* __restrict__ W, int nnz) {
    int i = blockIdx.x * blockDim.x + threadIdx.x;
    if (i < nnz) {
        atomicAdd(&W[(size_t)row_idx[i] * OUT_DIM + col_idx[i]], wv[i]);
    }
}

// ---------------------------------------------------------------------------
// Phase 3: f32 WMMA GEMM  out = x @ W + bias
//   256 threads (8 waves), block tile 128x128, K-stage 32.
//   LDS holds K-PAIR-packed tiles: A2[k2][m] = (A[m][2k2], A[m][2k2+1])
//   so every WMMA fragment is ONE ds_load_b64 into an adjacent VGPR pair.
//   Software-pipelined: global loads for stage s+1 issue before compute(s).
// ---------------------------------------------------------------------------
__global__ __launch_bounds__(256)
void wmma_gemm_bias_kernel(const float* __restrict__ X,
                           const float* __restrict__ W,
                           const float* __restrict__ bias,
                           float* __restrict__ out) {
    __shared__ float2 A2[KB / 2][ROWP];   // A2[k2][m ^ (k2<<1)]  (XOR-swizzled)
    __shared__ float2 B2[KB / 2][ROWP];   // B2[k2][n]            (no swizzle)

    const int tid   = threadIdx.x;
    const int lane  = tid & 31;
    const int wave  = tid >> 5;
    const int laneM = lane & 15;      // M (A frag) / N (B frag) index in tile
    const int h     = lane >> 4;      // 0: K={0,1}; 1: K={2,3} within a 4-step

    const int wm = (wave & 3) * 32;   // wave M offset in block tile
    const int wn = (wave >> 2) * 64;  // wave N offset in block tile
    const int blockM = blockIdx.y * BM;
    const int blockN = blockIdx.x * BN;

    // A-fragment swizzled column bases (h<<1 folded in; kk2<<1 XORed per step,
    // legal since kk2 is even so OR==ADD and bits stay disjoint)
    const int abase[2] = { (wm + 0  + laneM) ^ (h << 1),
                           (wm + 16 + laneM) ^ (h << 1) };

    // B staging: each thread owns one column, two K-rows per k2
    const int bcol  = tid & 127;
    const int bhalf = tid >> 7;

    v8f acc[2][4];
#pragma unroll
    for (int i = 0; i < 2; ++i)
#pragma unroll
        for (int j = 0; j < 4; ++j)
#pragma unroll
            for (int r = 0; r < 8; ++r) acc[i][j][r] = 0.f;

    float4 aReg[4];
    float  bReg0[8], bReg1[8];

    auto loadA = [&](int kb) {
#pragma unroll
        for (int t = 0; t < 4; ++t) {
            const int item = tid + t * 256;        // 1024 float4 items
            const int m    = item >> 3;            // 8 float4 per M-row
            const int c4   = (item & 7) << 2;      // K offset in stage
            aReg[t] = *reinterpret_cast<const float4*>(
                &X[(size_t)(blockM + m) * IN_DIM + kb + c4]);
        }
    };
    auto loadB = [&](int kb) {
#pragma unroll
        for (int r = 0; r < 8; ++r) {
            const int kr = kb + ((r * 2 + bhalf) << 1);   // = kb + 2*k2
            bReg0[r] = W[(size_t)(kr + 0) * OUT_DIM + blockN + bcol];
            bReg1[r] = W[(size_t)(kr + 1) * OUT_DIM + blockN + bcol];
        }
    };
    auto storeAB = [&]() {
#pragma unroll
        for (int t = 0; t < 4; ++t) {
            const int item = tid + t * 256;
            const int m    = item >> 3;
            const int k2   = (item & 7) << 1;
            A2[k2 + 0][m ^ ((k2 + 0) << 1)] = make_float2(aReg[t].x, aReg[t].y);
            A2[k2 + 1][m ^ ((k2 + 1) << 1)] = make_float2(aReg[t].z, aReg[t].w);
        }
#pragma unroll
        for (int r = 0; r < 8; ++r) {
            const int k2 = r * 2 + bhalf;
            B2[k2][bcol] = make_float2(bReg0[r], bReg1[r]);
        }
    };

    // prologue: stage 0 into LDS
    loadA(0);
    loadB(0);
    storeAB();

    for (int kb = 0; kb < IN_DIM; kb += KB) {
        __syncthreads();
        const bool more = (kb + KB) < IN_DIM;
        if (more) { loadA(kb + KB); loadB(kb + KB); }   // overlap with compute

#pragma unroll
        for (int kk2 = 0; kk2 < KB / 2; kk2 += 2) {     // 8 k-steps of K=4
            v2f a[2], b[4];
#pragma unroll
            for (int i = 0; i < 2; ++i)
                a[i] = *reinterpret_cast<const v2f*>(
                    &A2[kk2 + h][abase[i] ^ (kk2 << 1)]);
#pragma unroll
            for (int j = 0; j < 4; ++j)
                b[j] = *reinterpret_cast<const v2f*>(
                    &B2[kk2 + h][wn + j * 16 + laneM]);
#pragma unroll
            for (int i = 0; i < 2; ++i)
#pragma unroll
                for (int j = 0; j < 4; ++j)
                    acc[i][j] = __builtin_amdgcn_wmma_f32_16x16x4_f32(
                        false, a[i], false, b[j],
                        (short)0, acc[i][j], false, false);
        }

        __syncthreads();
        if (more) storeAB();
    }

    // epilogue: C/D layout (VGPR r -> rows r and r+8 per half-wave), add bias
    const int rowAdd = (lane >= 16) ? 8 : 0;
#pragma unroll
    for (int j = 0; j < 4; ++j) {
        const int col = blockN + wn + j * 16 + laneM;
        const float bv = bias[col];
#pragma unroll
        for (int i = 0; i < 2; ++i) {
            const int mbase = blockM + wm + i * 16 + rowAdd;
#pragma unroll
            for (int r = 0; r < 8; ++r) {
                out[(size_t)(mbase + r) * OUT_DIM + col] = acc[i][j][r] + bv;
            }
        }
    }
}

// ---------------------------------------------------------------------------
extern "C" void kernel_launch(void* const* d_in, const int* in_sizes, int n_in,
                              void* d_out, int out_size, void* d_ws, size_t ws_size,
                              hipStream_t stream) {
    const float* x       = (const float*)d_in[0];
    const int*   row_idx = (const int*)  d_in[1];
    const int*   col_idx = (const int*)  d_in[2];
    const float* weights = (const float*)d_in[3];
    const float* bias    = (const float*)d_in[4];
    float*       out     = (float*)d_out;
    float*       W       = (float*)d_ws;   // dense [IN_DIM, OUT_DIM] f32 = 128 MB

    (void)in_sizes; (void)n_in; (void)out_size; (void)ws_size;

    // 1) W = 0
    const int n4 = (IN_DIM * OUT_DIM) / 4;
    zero_w_kernel<<<(n4 + 255) / 256, 256, 0, stream>>>((float4*)W, n4);

    // 2) scatter-add COO entries
    scatter_w_kernel<<<(NNZ_CNT + 255) / 256, 256, 0, stream>>>(
        row_idx, col_idx, weights, W, NNZ_CNT);

    // 3) out = x @ W + bias  (f32 WMMA, software-pipelined)
    dim3 grid(OUT_DIM / BN, BATCH / BM);
    wmma_gemm_bias_kernel<<<grid, 256, 0, stream>>>(x, W, bias, out);
}